// RepulsionReppointsLoss_42228118454330
// MI455X (gfx1250) — compile-verified
//
#include <hip/hip_runtime.h>
#include <math.h>

#define PI_F 3.14159265358979323846f

typedef float v2f __attribute__((ext_vector_type(2)));
typedef float v8f __attribute__((ext_vector_type(8)));

__device__ __forceinline__ float norm_le135(float a) {
    float m = fmodf(a + PI_F * 0.25f, PI_F);
    if (m < 0.f) m += PI_F;
    return m - PI_F * 0.25f;
}

__device__ __forceinline__ float smooth_ln_f(float x) {
    const float LN2 = 0.69314718055994530942f;
    return (x <= 0.5f) ? -logf(1.0f - x) : (2.0f * (x - 0.5f) + LN2);
}

// ---------------------------------------------------------------------------
// WMMA block reduction: 256 f32 values (one per thread) staged in LDS as a
// 16x16 tile V (row-major). Wave 0 computes rowsum(V) with 4 chained
// V_WMMA_F32_16X16X4_F32 against an all-ones B, then folds the 16 row sums.
// A-matrix 16x4 f32 layout (ISA 7.12.2): lane l -> M=l&15; lanes 0-15 hold
// K=0,1 in VGPR0/1, lanes 16-31 hold K=2,3.
// Returns the total in every lane of wave 0.
// ---------------------------------------------------------------------------
__device__ __forceinline__ float wmma_tile_total(const float* tile, int tid) {
    int m  = tid & 15;
    int kb = (tid >> 4) << 1;   // 0 or 2
    v8f acc = {};
    v2f b;  b.x = 1.0f; b.y = 1.0f;
#pragma unroll
    for (int k = 0; k < 4; ++k) {
        v2f a;
        a.x = tile[m * 16 + 4 * k + kb];
        a.y = tile[m * 16 + 4 * k + kb + 1];
        acc = __builtin_amdgcn_wmma_f32_16x16x4_f32(false, a, false, b,
                                                    (short)0, acc, false, false);
    }
    // D[v][lane]: lanes<16 cover M=v (0..7), lanes>=16 cover M=v+8 (8..15)
    float s = acc[0] + acc[1] + acc[2] + acc[3] + acc[4] + acc[5] + acc[6] + acc[7];
    s += __shfl_xor(s, 16, 32);
    return s;
}

// ---------------------------------------------------------------------------
// Exact replica of reference quad_intersection_area for one pair of quads.
// q1,q2: 8 floats each [x0,y0,x1,y1,x2,y2,x3,y3].
// ---------------------------------------------------------------------------
__device__ float quad_inter_area(const float* __restrict__ q1,
                                 const float* __restrict__ q2) {
    float c1x[4], c1y[4], c2x[4], c2y[4];
#pragma unroll
    for (int k = 0; k < 4; ++k) {
        c1x[k] = q1[2 * k]; c1y[k] = q1[2 * k + 1];
        c2x[k] = q2[2 * k]; c2y[k] = q2[2 * k + 1];
    }

    float vx[24], vy[24];
    bool  mk[24];

    // c1 corners inside rect c2 -> slots 0..3
    {
        float abx = c2x[1] - c2x[0], aby = c2y[1] - c2y[0];
        float adx = c2x[3] - c2x[0], ady = c2y[3] - c2y[0];
        float ab2 = abx * abx + aby * aby, ad2 = adx * adx + ady * ady;
#pragma unroll
        for (int k = 0; k < 4; ++k) {
            vx[k] = c1x[k]; vy[k] = c1y[k];
            float apx = c1x[k] - c2x[0], apy = c1y[k] - c2y[0];
            float pab = apx * abx + apy * aby, pad = apx * adx + apy * ady;
            mk[k] = (pab >= 0.f) && (pab <= ab2) && (pad >= 0.f) && (pad <= ad2);
        }
    }
    // c2 corners inside rect c1 -> slots 4..7
    {
        float abx = c1x[1] - c1x[0], aby = c1y[1] - c1y[0];
        float adx = c1x[3] - c1x[0], ady = c1y[3] - c1y[0];
        float ab2 = abx * abx + aby * aby, ad2 = adx * adx + ady * ady;
#pragma unroll
        for (int k = 0; k < 4; ++k) {
            vx[4 + k] = c2x[k]; vy[4 + k] = c2y[k];
            float apx = c2x[k] - c1x[0], apy = c2y[k] - c1y[0];
            float pab = apx * abx + apy * aby, pad = apx * adx + apy * ady;
            mk[4 + k] = (pab >= 0.f) && (pab <= ab2) && (pad >= 0.f) && (pad <= ad2);
        }
    }
    // edge-edge intersections -> slots 8..23 (row-major i*4+j, matches ref)
#pragma unroll
    for (int i = 0; i < 4; ++i) {
        float p1x = c1x[i], p1y = c1y[i];
        float d21x = c1x[(i + 1) & 3] - p1x, d21y = c1y[(i + 1) & 3] - p1y;
#pragma unroll
        for (int j = 0; j < 4; ++j) {
            float p3x = c2x[j], p3y = c2y[j];
            float d43x = c2x[(j + 1) & 3] - p3x, d43y = c2y[(j + 1) & 3] - p3y;
            float d31x = p3x - p1x, d31y = p3y - p1y;
            float den  = d21x * d43y - d21y * d43x;
            float safe = (den == 0.f) ? 1.f : den;
            float t = (d31x * d43y - d31y * d43x) / safe;
            float u = (d31x * d21y - d31y * d21x) / safe;
            bool hit = (den != 0.f) && (t > 0.f) && (t < 1.f) && (u > 0.f) && (u < 1.f);
            int s = 8 + i * 4 + j;
            vx[s] = p1x + t * d21x;
            vy[s] = p1y + t * d21y;
            mk[s] = hit;
        }
    }
    // masked mean
    float sx = 0.f, sy = 0.f; int num = 0;
#pragma unroll
    for (int k = 0; k < 24; ++k)
        if (mk[k]) { sx += vx[k]; sy += vy[k]; ++num; }
    float inv = 1.0f / (float)(num > 0 ? num : 1);
    float mx = sx * inv, my = sy * inv;

    float ang[24];
#pragma unroll
    for (int k = 0; k < 24; ++k) {
        vx[k] -= mx; vy[k] -= my;
        ang[k] = mk[k] ? atan2f(vy[k], vx[k]) : 1.0e6f;
    }

    // global minimum key (angle, index) among valid — strict < keeps first
    float gA = 3.0e38f, gX = 0.f, gY = 0.f;
#pragma unroll
    for (int j = 0; j < 24; ++j)
        if (mk[j] && (ang[j] < gA)) { gA = ang[j]; gX = vx[j]; gY = vy[j]; }

    // shoelace over angle-sorted valid verts via successor scan
    // (equivalent to reference's stable argsort + roll)
    float cr = 0.f;
#pragma unroll
    for (int k = 0; k < 24; ++k) {
        if (!mk[k]) continue;
        float bA = 3.0e38f; int bI = 10000;
        float bX = gX, bY = gY;   // wrap to global-min if no successor
#pragma unroll
        for (int j = 0; j < 24; ++j) {
            if (!mk[j]) continue;
            bool gt = (ang[j] > ang[k]) || ((ang[j] == ang[k]) && (j > k));
            bool lt = (ang[j] < bA) || ((ang[j] == bA) && (j < bI));
            if (gt && lt) { bA = ang[j]; bI = j; bX = vx[j]; bY = vy[j]; }
        }
        cr += vx[k] * bY - vy[k] * bX;
    }
    return 0.5f * fabsf(cr);
}

__device__ __forceinline__ void write_corners(float cx, float cy, float w, float h,
                                              float a, float* out) {
    const float xs[4] = {0.5f, -0.5f, -0.5f, 0.5f};
    const float ys[4] = {0.5f, 0.5f, -0.5f, -0.5f};
    float ca = cosf(a), sa = sinf(a);
#pragma unroll
    for (int k = 0; k < 4; ++k) {
        float x4 = w * xs[k], y4 = h * ys[k];
        out[2 * k]     = x4 * ca - y4 * sa + cx;
        out[2 * k + 1] = x4 * sa + y4 * ca + cy;
    }
}

__device__ __forceinline__ void write_xyxy(float cx, float cy, float w, float h,
                                           float a, float* out) {
    float ca = fabsf(cosf(a)), sa = fabsf(sinf(a));
    float hw = (ca * w + sa * h) * 0.5f;
    float hh = (sa * w + ca * h) * 0.5f;
    out[0] = cx - hw; out[1] = cy - hh; out[2] = cx + hw; out[3] = cy + hh;
}

// ---------------------------------------------------------------------------
// Accumulator slots in ws:  [0]=sum_repgt f32  [1]=n_gt u32  [2]=sum_repbox
// [3]=n_box u32  [4]=n_pos u32  [5]=sum_attr f32  [6]=n_rep u32
// ---------------------------------------------------------------------------
__global__ void k_init(float* W) {
    if (threadIdx.x < 16) W[threadIdx.x] = 0.f;
}

__global__ void k_prep(const float* __restrict__ pts, const float* __restrict__ bgt,
                       const int* __restrict__ labels, const int* __restrict__ pos,
                       int P, float* predObb, float* gtObb, float* pc, float* gc,
                       float* pxy, float* gxy, float* grows, int* lab) {
    int t = blockIdx.x * blockDim.x + threadIdx.x;
    if (t >= P) return;
    int idx = pos[t];

    // ---- min_area_obb over 9 gathered points ----
    float px[9], py[9];
#pragma unroll
    for (int k = 0; k < 9; ++k) {
        px[k] = pts[idx * 18 + 2 * k];
        py[k] = pts[idx * 18 + 2 * k + 1];
    }
    float bestA = 3.0e38f, bw = 0, bh = 0, bcx = 0, bcy = 0, bc = 1, bs = 0, bth = 0;
    for (int i = 0; i < 9; ++i) {
        for (int j = i + 1; j < 9; ++j) {            // triu order == ref argmin order
            float ang = atan2f(py[i] - py[j], px[i] - px[j]);
            float c = cosf(ang), s = sinf(ang);
            float xmin = 3e38f, xmax = -3e38f, ymin = 3e38f, ymax = -3e38f;
#pragma unroll
            for (int k = 0; k < 9; ++k) {
                float x =  px[k] * c + py[k] * s;
                float y = -px[k] * s + py[k] * c;
                xmin = fminf(xmin, x); xmax = fmaxf(xmax, x);
                ymin = fminf(ymin, y); ymax = fmaxf(ymax, y);
            }
            float area = (xmax - xmin) * (ymax - ymin);
            if (area < bestA) {                       // strict < == first argmin
                bestA = area;
                bw = xmax - xmin; bh = ymax - ymin;
                bcx = (xmax + xmin) * 0.5f; bcy = (ymax + ymin) * 0.5f;
                bc = c; bs = s; bth = ang;
            }
        }
    }
    float cx = bcx * bc - bcy * bs;
    float cy = bcx * bs + bcy * bc;
    bool  sw = bh > bw;
    float ww = sw ? bh : bw;
    float hh = sw ? bw : bh;
    float th = norm_le135(sw ? bth + PI_F * 0.5f : bth);
    predObb[t * 5 + 0] = cx; predObb[t * 5 + 1] = cy;
    predObb[t * 5 + 2] = ww; predObb[t * 5 + 3] = hh; predObb[t * 5 + 4] = th;
    write_corners(cx, cy, ww, hh, th, pc + (size_t)t * 8);
    write_xyxy(cx, cy, ww, hh, th, pxy + (size_t)t * 4);

    // ---- poly2obb for gathered gt ----
    float g[8];
#pragma unroll
    for (int k = 0; k < 8; ++k) { g[k] = bgt[idx * 8 + k]; grows[t * 8 + k] = g[k]; }
    lab[t] = labels[idx];
    float e1 = sqrtf((g[0] - g[2]) * (g[0] - g[2]) + (g[1] - g[3]) * (g[1] - g[3]));
    float e2 = sqrtf((g[2] - g[4]) * (g[2] - g[4]) + (g[3] - g[5]) * (g[3] - g[5]));
    float a1 = atan2f(g[3] - g[1], g[2] - g[0]);
    float a2 = atan2f(g[7] - g[1], g[6] - g[0]);
    float ga = norm_le135((e1 > e2) ? a1 : a2);
    float gcx = (g[0] + g[4]) * 0.5f, gcy = (g[1] + g[5]) * 0.5f;
    float gw = fmaxf(e1, e2), gh = fminf(e1, e2);
    gtObb[t * 5 + 0] = gcx; gtObb[t * 5 + 1] = gcy;
    gtObb[t * 5 + 2] = gw;  gtObb[t * 5 + 3] = gh; gtObb[t * 5 + 4] = ga;
    write_corners(gcx, gcy, gw, gh, ga, gc + (size_t)t * 8);
    write_xyxy(gcx, gcy, gw, gh, ga, gxy + (size_t)t * 4);
}

__global__ void k_diag(const float* predObb, const float* gtObb,
                       const float* pc, const float* gc, int P, float* iouself) {
    int t = blockIdx.x * blockDim.x + threadIdx.x;
    if (t >= P) return;
    float inter = quad_inter_area(pc + (size_t)t * 8, gc + (size_t)t * 8);
    float u = predObb[t * 5 + 2] * predObb[t * 5 + 3] +
              gtObb[t * 5 + 2] * gtObb[t * 5 + 3] - inter;
    iouself[t] = inter / fmaxf(u, 1e-8f);
}

__global__ void k_attr(const float* __restrict__ grows,
                       const float* __restrict__ iouself, int P, float* W) {
    __shared__ float ssum[256];
    __shared__ int   scnt;
    int tid = threadIdx.x;
    if (tid == 0) scnt = 0;
    __syncthreads();

    int i = blockIdx.x * blockDim.x + tid;
    float contrib = 0.f;
    if (i < P) {
        float g[8];
#pragma unroll
        for (int k = 0; k < 8; ++k) g[k] = grows[i * 8 + k];
        int first = -1; float gmax = -3.0e38f;
        for (int j = 0; j < P; ++j) {
            bool same = true;
#pragma unroll
            for (int k = 0; k < 8; ++k) same = same && (g[k] == grows[j * 8 + k]);
            if (same) {
                if (first < 0) first = j;
                gmax = fmaxf(gmax, iouself[j]);
            }
        }
        if (first == i) { contrib = 1.0f - gmax; atomicAdd(&scnt, 1); }
    }
    ssum[tid] = contrib;
    __syncthreads();
    for (int s = 128; s > 0; s >>= 1) {
        if (tid < s) ssum[tid] += ssum[tid + s];
        __syncthreads();
    }
    if (tid == 0) {
        atomicAdd(&W[5], ssum[0]);
        atomicAdd(((unsigned int*)W) + 6, (unsigned int)scnt);
    }
}

__global__ void k_pg(const float* __restrict__ grows, const int* __restrict__ lab,
                     const float* __restrict__ pxy, const float* __restrict__ gxy,
                     const float* __restrict__ gtObb, const float* __restrict__ pc,
                     const float* __restrict__ gc, int P, float* W) {
    __shared__ float tile[256];
    __shared__ int   cnt;
    int tid = threadIdx.x;
    if (tid == 0) cnt = 0;
    __syncthreads();

    long long idx = (long long)blockIdx.x * 256 + tid;
    long long tot = (long long)P * P;
    float val = 0.f;
    if (idx < tot) {
        int i = (int)(idx / P), j = (int)(idx % P);
        if (i != j && lab[i] == lab[j]) {
            bool diff = false;
#pragma unroll
            for (int k = 0; k < 8; ++k) diff = diff || (grows[i * 8 + k] != grows[j * 8 + k]);
            if (diff) {
                float ax0 = pxy[i * 4], ay0 = pxy[i * 4 + 1];
                float ax1 = pxy[i * 4 + 2], ay1 = pxy[i * 4 + 3];
                float bx0 = gxy[j * 4], by0 = gxy[j * 4 + 1];
                float bx1 = gxy[j * 4 + 2], by1 = gxy[j * 4 + 3];
                float iw = fmaxf(fminf(ax1, bx1) - fmaxf(ax0, bx0), 0.f);
                float ih = fmaxf(fminf(ay1, by1) - fmaxf(ay0, by0), 0.f);
                float ua = fmaxf((ax1 - ax0) * (ay1 - ay0) +
                                 (bx1 - bx0) * (by1 - by0) - iw * ih, 1e-8f);
                if (iw * ih / ua > 0.f) {
                    float inter = quad_inter_area(pc + (size_t)i * 8, gc + (size_t)j * 8);
                    float iog = inter / (gtObb[j * 5 + 2] * gtObb[j * 5 + 3]);
                    val = smooth_ln_f(iog);
                    atomicAdd(&cnt, 1);
                }
            }
        }
    }
    tile[tid] = val;
    __syncthreads();
    if (tid < 32) {
        float s = wmma_tile_total(tile, tid);       // v_wmma_f32_16x16x4_f32 x4
        if (tid == 0) {
            atomicAdd(&W[0], s);
            atomicAdd(((unsigned int*)W) + 1, (unsigned int)cnt);
        }
    }
}

__global__ void k_pp(const float* __restrict__ grows, const int* __restrict__ lab,
                     const float* __restrict__ pxy, const float* __restrict__ predObb,
                     const float* __restrict__ pc, int P, float* W) {
    __shared__ float tile[256];
    __shared__ int   cntBox;
    __shared__ int   cntPos;
    int tid = threadIdx.x;
    if (tid == 0) { cntBox = 0; cntPos = 0; }
    __syncthreads();

    long long idx = (long long)blockIdx.x * 256 + tid;
    long long tot = (long long)P * P;
    float val = 0.f;
    if (idx < tot) {
        int i = (int)(idx / P), j = (int)(idx % P);
        if (j > i && lab[i] == lab[j]) {
            bool diff = false;
#pragma unroll
            for (int k = 0; k < 8; ++k) diff = diff || (grows[i * 8 + k] != grows[j * 8 + k]);
            if (diff) {
                float ax0 = pxy[i * 4], ay0 = pxy[i * 4 + 1];
                float ax1 = pxy[i * 4 + 2], ay1 = pxy[i * 4 + 3];
                float bx0 = pxy[j * 4], by0 = pxy[j * 4 + 1];
                float bx1 = pxy[j * 4 + 2], by1 = pxy[j * 4 + 3];
                float iw = fmaxf(fminf(ax1, bx1) - fmaxf(ax0, bx0), 0.f);
                float ih = fmaxf(fminf(ay1, by1) - fmaxf(ay0, by0), 0.f);
                float ua = fmaxf((ax1 - ax0) * (ay1 - ay0) +
                                 (bx1 - bx0) * (by1 - by0) - iw * ih, 1e-8f);
                if (iw * ih / ua > 0.f) {
                    float inter = quad_inter_area(pc + (size_t)i * 8, pc + (size_t)j * 8);
                    float un = predObb[i * 5 + 2] * predObb[i * 5 + 3] +
                               predObb[j * 5 + 2] * predObb[j * 5 + 3] - inter;
                    float iou = inter / fmaxf(un, 1e-8f);
                    val = smooth_ln_f(iou);
                    atomicAdd(&cntBox, 1);
                    if (iou > 0.f) atomicAdd(&cntPos, 1);
                }
            }
        }
    }
    tile[tid] = val;
    __syncthreads();
    if (tid < 32) {
        float s = wmma_tile_total(tile, tid);       // v_wmma_f32_16x16x4_f32 x4
        if (tid == 0) {
            atomicAdd(&W[2], s);
            atomicAdd(((unsigned int*)W) + 3, (unsigned int)cntBox);
            atomicAdd(((unsigned int*)W) + 4, (unsigned int)cntPos);
        }
    }
}

__global__ void k_final(const float* W, float* out) {
    const unsigned int* U = (const unsigned int*)W;
    float sum_gt  = W[0]; unsigned n_gt  = U[1];
    float sum_box = W[2]; unsigned n_box = U[3], n_pos = U[4];
    float sum_attr = W[5]; unsigned n_rep = U[6];
    out[0] = sum_attr / (float)(n_rep > 0 ? n_rep : 1);                       // loss_attr
    out[1] = 0.5f * ((n_gt  > 0) ? sum_gt / (float)n_gt : 0.f);               // loss_repgt
    out[2] = 0.5f * ((n_box > 0) ? (sum_box / (float)n_box) / fmaxf((float)n_pos, 1.f)
                                 : 0.f);                                      // loss_repbox
}

extern "C" void kernel_launch(void* const* d_in, const int* in_sizes, int n_in,
                              void* d_out, int out_size, void* d_ws, size_t ws_size,
                              hipStream_t stream) {
    const float* pts    = (const float*)d_in[0];
    const float* bgt    = (const float*)d_in[1];
    const int*   labels = (const int*)d_in[2];
    const int*   pos    = (const int*)d_in[3];
    int P = in_sizes[3];

    float* W       = (float*)d_ws;
    float* predObb = W + 16;
    float* gtObb   = predObb + (size_t)5 * P;
    float* pc      = gtObb   + (size_t)5 * P;
    float* gc      = pc      + (size_t)8 * P;
    float* pxy     = gc      + (size_t)8 * P;
    float* gxy     = pxy     + (size_t)4 * P;
    float* grows   = gxy     + (size_t)4 * P;
    int*   lab     = (int*)(grows + (size_t)8 * P);
    float* iouself = (float*)(lab + P);

    k_init<<<1, 64, 0, stream>>>(W);
    int pb = (P + 255) / 256;
    k_prep<<<pb, 256, 0, stream>>>(pts, bgt, labels, pos, P, predObb, gtObb,
                                   pc, gc, pxy, gxy, grows, lab);
    k_diag<<<pb, 256, 0, stream>>>(predObb, gtObb, pc, gc, P, iouself);
    k_attr<<<pb, 256, 0, stream>>>(grows, iouself, P, W);

    long long tot = (long long)P * P;
    int gb = (int)((tot + 255) / 256);
    k_pg<<<gb, 256, 0, stream>>>(grows, lab, pxy, gxy, gtObb, pc, gc, P, W);
    k_pp<<<gb, 256, 0, stream>>>(grows, lab, pxy, predObb, pc, P, W);

    k_final<<<1, 1, 0, stream>>>(W, (float*)d_out);
}